// RCNNHead_59081570124838
// MI455X (gfx1250) — compile-verified
//
#include <hip/hip_runtime.h>
#include <math.h>
#include <stdint.h>

// ---------------- problem constants (match reference) ----------------
#define B_ 8
#define M_ 1024
#define N_ 100
#define ROI_PER_IMAGE 128
#define FG_PER_IMAGE 64
#define REG_FG_THRESH 0.55f
#define CLS_FG_THRESH 0.75f
#define CLS_BG_THRESH 0.25f
#define CLS_BG_THRESH_LO 0.1f
#define HARD_BG_RATIO 0.8f
// FG_THRESH = min(REG_FG_THRESH, CLS_FG_THRESH) = 0.55

// ---------------- d_out layout (floats, concatenated in return order) ----
#define OFF_ROIS 0
#define OFF_GT   (B_ * ROI_PER_IMAGE * 7)                  // 7168
#define OFF_IOU  (OFF_GT + B_ * ROI_PER_IMAGE * 8)         // 15360
#define OFF_SC   (OFF_IOU + B_ * ROI_PER_IMAGE)            // 16384
#define OFF_LB   (OFF_SC  + B_ * ROI_PER_IMAGE)            // 17408
#define OFF_CLS  (OFF_LB  + B_ * ROI_PER_IMAGE)            // 18432
#define OFF_RV   (OFF_CLS + B_ * ROI_PER_IMAGE)            // 19456

// ---------------- small utils ----------------
__device__ __forceinline__ int imin_(int a, int b) { return a < b ? a : b; }

__device__ __forceinline__ uint32_t hash_u32(uint32_t x) {
  x ^= x >> 16; x *= 0x7feb352dU;
  x ^= x >> 15; x *= 0x846ca68bU;
  x ^= x >> 16; return x;
}
// deterministic per-(stream, image, slot) uniform in [0,1)
__device__ __forceinline__ float urand(uint32_t stream, uint32_t b, uint32_t i) {
  uint32_t h = hash_u32(hash_u32(i * 0x9E3779B9u + b * 0x85ebca6bu + stream * 0xC2B2AE35u)
                        + 0x27d4eb2fu);
  return (float)(h >> 8) * (1.0f / 16777216.0f);
}

#define INVALID_KEY 0xFFFFFFFFFFFFFFFFull

// ---------------- rotated 3D IoU of one (roi, gt) pair ----------------
// Faithful port of _pair_iou3d. All polygon state is kept in registers:
// the 24 candidate vertices (+8 sentinel pads) are sorted by angle with a
// fully-unrolled 32-element bitonic network over (u64 key, x, y) triples,
// so there is no scratch spill in the hot loop.
__device__ __forceinline__ float pair_iou3d(const float a[7], const float* __restrict__ bq) {
  const float EPS = 1e-8f;
  float cax[4], cay[4], cbx[4], cby[4];
  const float LX[4] = {1.f, 1.f, -1.f, -1.f};
  const float LY[4] = {1.f, -1.f, -1.f, 1.f};
  float ca = cosf(a[6]), sa = sinf(a[6]);
  float cb = cosf(bq[6]), sb = sinf(bq[6]);
  {
    float hx = a[3] * 0.5f, hy = a[4] * 0.5f;
#pragma unroll
    for (int i = 0; i < 4; ++i) {
      float X = LX[i] * hx, Y = LY[i] * hy;
      cax[i] = X * ca - Y * sa + a[0];
      cay[i] = X * sa + Y * ca + a[1];
    }
  }
  {
    float hx = bq[3] * 0.5f, hy = bq[4] * 0.5f;
#pragma unroll
    for (int i = 0; i < 4; ++i) {
      float X = LX[i] * hx, Y = LY[i] * hy;
      cbx[i] = X * cb - Y * sb + bq[0];
      cby[i] = X * sb + Y * cb + bq[1];
    }
  }

  float xx[32], yy[32];
  unsigned vmask = 0;  // bits 0..23: candidate validity

  // 16 edge-edge crossing candidates (order = i*4 + j, matching reshape(16,2))
#pragma unroll
  for (int i = 0; i < 4; ++i) {
    float a1x = cax[i], a1y = cay[i];
    float a2x = cax[(i + 1) & 3], a2y = cay[(i + 1) & 3];
    float d1x = a2x - a1x, d1y = a2y - a1y;
#pragma unroll
    for (int j = 0; j < 4; ++j) {
      float b1x = cbx[j], b1y = cby[j];
      float b2x = cbx[(j + 1) & 3], b2y = cby[(j + 1) & 3];
      float d2x = b2x - b1x, d2y = b2y - b1y;
      float dfx = b1x - a1x, dfy = b1y - a1y;
      float den = d1x * d2y - d1y * d2x;
      bool nz = fabsf(den) > EPS;
      float safe = nz ? den : 1.0f;
      float t = (dfx * d2y - dfy * d2x) / safe;
      float u = (dfx * d1y - dfy * d1x) / safe;
      int q = i * 4 + j;
      if (nz && (t >= 0.f) && (t <= 1.f) && (u >= 0.f) && (u <= 1.f)) vmask |= (1u << q);
      xx[q] = a1x + t * d1x;
      yy[q] = a1y + t * d1y;
    }
  }
  // corners of A inside B
#pragma unroll
  for (int i = 0; i < 4; ++i) {
    float rx0 = cax[i] - bq[0], ry0 = cay[i] - bq[1];
    float lx = rx0 * cb + ry0 * sb;
    float ly = -rx0 * sb + ry0 * cb;
    if ((fabsf(lx) <= bq[3] * 0.5f + 1e-6f) && (fabsf(ly) <= bq[4] * 0.5f + 1e-6f))
      vmask |= (1u << (16 + i));
    xx[16 + i] = cax[i]; yy[16 + i] = cay[i];
  }
  // corners of B inside A
#pragma unroll
  for (int i = 0; i < 4; ++i) {
    float rx0 = cbx[i] - a[0], ry0 = cby[i] - a[1];
    float lx = rx0 * ca + ry0 * sa;
    float ly = -rx0 * sa + ry0 * ca;
    if ((fabsf(lx) <= a[3] * 0.5f + 1e-6f) && (fabsf(ly) <= a[4] * 0.5f + 1e-6f))
      vmask |= (1u << (20 + i));
    xx[20 + i] = cbx[i]; yy[20 + i] = cby[i];
  }

  // masked centroid
  int nv = 0;
  float sx = 0.f, sy = 0.f;
#pragma unroll
  for (int q = 0; q < 24; ++q) {
    if ((vmask >> q) & 1u) { ++nv; sx += xx[q]; sy += yy[q]; }
  }
  float inv = 1.0f / fmaxf((float)nv, 1.0f);
  float ctx = sx * inv, cty = sy * inv;

  // sort keys: order-preserving uint encoding of angle, index in low 5 bits
  // (unique keys -> equivalent to jnp's stable argsort). Invalid -> max key.
  unsigned long long key[32];
#pragma unroll
  for (int q = 0; q < 24; ++q) {
    if ((vmask >> q) & 1u) {
      float angv = atan2f(yy[q] - cty, xx[q] - ctx);
      unsigned u = __float_as_uint(angv);
      unsigned ord = (u & 0x80000000u) ? ~u : (u | 0x80000000u);
      key[q] = (((unsigned long long)ord) << 5) | (unsigned)q;
    } else {
      key[q] = INVALID_KEY;
    }
  }
#pragma unroll
  for (int q = 24; q < 32; ++q) { key[q] = INVALID_KEY; xx[q] = 0.f; yy[q] = 0.f; }

  // 32-element bitonic sorting network, fully unrolled -> register resident
#define CSWAP_(i1, i2, up_)                                                    \
  {                                                                            \
    bool sw_ = (key[i1] > key[i2]) == (up_);                                   \
    unsigned long long kt_ = key[i1];                                          \
    key[i1] = sw_ ? key[i2] : kt_;                                             \
    key[i2] = sw_ ? kt_ : key[i2];                                             \
    float xt_ = xx[i1];                                                        \
    xx[i1] = sw_ ? xx[i2] : xt_;                                               \
    xx[i2] = sw_ ? xt_ : xx[i2];                                               \
    float yt_ = yy[i1];                                                        \
    yy[i1] = sw_ ? yy[i2] : yt_;                                               \
    yy[i2] = sw_ ? yt_ : yy[i2];                                               \
  }
#pragma unroll
  for (int kk = 2; kk <= 32; kk <<= 1) {
#pragma unroll
    for (int j = kk >> 1; j > 0; j >>= 1) {
#pragma unroll
      for (int i = 0; i < 32; ++i) {
        int ix = i ^ j;
        if (ix > i) {
          CSWAP_(i, ix, ((i & kk) == 0));
        }
      }
    }
  }
#undef CSWAP_

  // invalid (and pad) entries -> duplicate first vertex: zero area contribution
  bool v0 = key[0] != INVALID_KEY;
  float f0x = v0 ? (xx[0] - ctx) : 0.f;
  float f0y = v0 ? (yy[0] - cty) : 0.f;
  float acc = 0.f;
  float prevx = f0x, prevy = f0y;
#pragma unroll
  for (int k = 1; k < 32; ++k) {
    bool vk = key[k] != INVALID_KEY;
    float vx = vk ? (xx[k] - ctx) : f0x;
    float vy = vk ? (yy[k] - cty) : f0y;
    acc += prevx * vy - prevy * vx;
    prevx = vx; prevy = vy;
  }
  acc += prevx * f0y - prevy * f0x;  // close the ring
  float area = 0.5f * fabsf(acc);

  float top = fminf(a[2] + a[5] * 0.5f, bq[2] + bq[5] * 0.5f);
  float bot = fmaxf(a[2] - a[5] * 0.5f, bq[2] - bq[5] * 0.5f);
  float oh = fmaxf(top - bot, 0.f);
  float inter = area * oh;
  float va = a[3] * a[4] * a[5];
  float vb = bq[3] * bq[4] * bq[5];
  return inter / fmaxf(va + vb - inter, 1e-6f);
}

// ---------------- phase 1: IoU max/argmax per ROI ----------------
// grid = (B, M/128), block = 128 (4 wave32 per block).
// GT table (800 dwords) staged to LDS with CDNA5 async global->LDS loads.
__global__ __launch_bounds__(128) void rcnn_iou_kernel(
    const float* __restrict__ rois, const float* __restrict__ gt,
    float* __restrict__ mo, int* __restrict__ asgn) {
  const int b = blockIdx.x;
  __shared__ float sgt[N_ * 8];
  __shared__ int s_last;
  if (threadIdx.x == 0) s_last = -1;

  const int m = blockIdx.y * blockDim.x + threadIdx.x;  // [0, 1024)
  const float* rp = rois + ((size_t)b * M_ + m) * 7;
  __builtin_prefetch(rp, 0, 0);  // global_prefetch_b8 for this lane's ROI row

  // Async-copy gt[b] (N*8 dwords) into LDS: ASYNCcnt-tracked DMA-style loads.
  const float* g = gt + (size_t)b * N_ * 8;
  for (int i = threadIdx.x; i < N_ * 8; i += blockDim.x) {
    uint32_t lofs = (uint32_t)(uintptr_t)(&sgt[i]);   // low 32 bits of flat = LDS byte offset
    uint64_t ga = (uint64_t)(uintptr_t)(g + i);
    asm volatile("global_load_async_to_lds_b32 %0, %1, off"
                 :: "v"(lofs), "v"(ga) : "memory");
  }
#if __has_builtin(__builtin_amdgcn_s_wait_asynccnt)
  __builtin_amdgcn_s_wait_asynccnt(0);
#else
  asm volatile("s_wait_asynccnt 0x0" ::: "memory");
#endif
  __syncthreads();

  // last gt row with nonzero sum (mirrors the while-loop trim)
  for (int n = threadIdx.x; n < N_; n += blockDim.x) {
    float s = 0.f;
    for (int k = 0; k < 8; ++k) s += sgt[n * 8 + k];
    if (s != 0.f) atomicMax(&s_last, n);
  }
  __syncthreads();
  const int last = s_last;

  float a[7];
#pragma unroll
  for (int k = 0; k < 7; ++k) a[k] = rp[k];
  float best = -1.0f;  // invalid columns contribute -1.0 as in reference
  int bi = 0;
  for (int n = 0; n <= last; ++n) {
    float v = pair_iou3d(a, &sgt[n * 8]);
    if (v > best) { best = v; bi = n; }  // strict '>' => first argmax, like jnp
  }
  mo[b * M_ + m] = best;
  asgn[b * M_ + m] = bi;
}

// ---------------- phase 2: fg/bg sampling + gather + targets ----------------
// grid = B, block = 256 (8 wave32). All list building done in LDS.
__global__ __launch_bounds__(256) void rcnn_sample_kernel(
    const float* __restrict__ rois, const float* __restrict__ gt,
    const float* __restrict__ scores, const int* __restrict__ labels,
    const float* __restrict__ mo_g, const int* __restrict__ asgn_g,
    float* __restrict__ out) {
  const int b = blockIdx.x;
  const int tid = threadIdx.x;

  __shared__ float mo_s[M_];
  __shared__ int fg_asc[M_], hard_asc[M_], easy_asc[M_];
  __shared__ uint32_t keys[M_];
  __shared__ int vals[M_];
  __shared__ int scan_buf[256];
  __shared__ int counts[3];

  for (int i = tid; i < M_; i += 256) mo_s[i] = mo_g[b * M_ + i];
  __syncthreads();

  // Build fg/hard/easy ascending lists == argsort(where(mask, ar, M+ar)).
  for (int sel = 0; sel < 3; ++sel) {
    int* asc = (sel == 0) ? fg_asc : (sel == 1) ? hard_asc : easy_asc;
    const int base = tid * 4;
    int m4[4], sum = 0;
#pragma unroll
    for (int r = 0; r < 4; ++r) {
      float v = mo_s[base + r];
      bool fgm = v >= REG_FG_THRESH;        // FG_THRESH = 0.55
      bool ezm = v < CLS_BG_THRESH_LO;      // < 0.1
      bool hdm = !fgm && !ezm;              // [0.1, 0.55)
      m4[r] = (sel == 0) ? (int)fgm : (sel == 1) ? (int)hdm : (int)ezm;
      sum += m4[r];
    }
    scan_buf[tid] = sum;
    __syncthreads();
    for (int off = 1; off < 256; off <<= 1) {  // Hillis-Steele inclusive scan
      int add = (tid >= off) ? scan_buf[tid - off] : 0;
      __syncthreads();
      scan_buf[tid] += add;
      __syncthreads();
    }
    const int total = scan_buf[255];
    int t = scan_buf[tid] - sum;   // exclusive true-rank at chunk start
    int f = base - t;              // exclusive false-rank at chunk start
    if (tid == 0) counts[sel] = total;
#pragma unroll
    for (int r = 0; r < 4; ++r) {
      if (m4[r]) asc[t++] = base + r;
      else       asc[total + (f++)] = base + r;
    }
    __syncthreads();
  }

  // fg_perm: random permutation restricted to fg indices (fg first) via
  // bitonic sort of per-index random keys (non-fg => UINT_MAX, sorts last).
  for (int i = tid; i < M_; i += 256) {
    bool fgm = mo_s[i] >= REG_FG_THRESH;
    keys[i] = fgm ? hash_u32(i * 0x9E3779B9u + b * 0x85ebca6bu + 0x1b873593u) : 0xFFFFFFFFu;
    vals[i] = i;
  }
  __syncthreads();
  for (int ksz = 2; ksz <= M_; ksz <<= 1) {
    for (int j = ksz >> 1; j > 0; j >>= 1) {
      for (int i = tid; i < M_; i += 256) {
        int ixj = i ^ j;
        if (ixj > i) {
          bool up = ((i & ksz) == 0);
          uint32_t ka = keys[i], kc = keys[ixj];
          if ((ka > kc) == up) {
            keys[i] = kc; keys[ixj] = ka;
            int tv = vals[i]; vals[i] = vals[ixj]; vals[ixj] = tv;
          }
        }
      }
      __syncthreads();
    }
  }

  const int n_fg = counts[0], n_hard = counts[1], n_easy = counts[2];
  const int n_bg = n_hard + n_easy;
  const int fg_this = (n_fg > 0) ? ((n_bg > 0) ? imin_(FG_PER_IMAGE, n_fg) : ROI_PER_IMAGE) : 0;
  const int bg_this = ROI_PER_IMAGE - fg_this;
  const int hard_num = (n_hard > 0 && n_easy > 0) ? (int)((float)bg_this * HARD_BG_RATIO)
                     : ((n_hard > 0) ? bg_this : 0);

  if (tid < ROI_PER_IMAGE) {
    const int j = tid;
    // rpick(u, asc, n) = asc[clip(int(u*n), 0, max(n-1,0))]
    auto rpick = [](float u, const int* asc, int n) {
      int i = (int)(u * (float)n);
      int hi = (n - 1 > 0) ? (n - 1) : 0;
      i = (i < 0) ? 0 : ((i > hi) ? hi : i);
      return asc[i];
    };
    const int rand_fg = rpick(urand(2u, (uint32_t)b, (uint32_t)j), fg_asc, n_fg);
    const int hp      = rpick(urand(3u, (uint32_t)b, (uint32_t)j), hard_asc, n_hard);
    const int ep      = rpick(urand(4u, (uint32_t)b, (uint32_t)j), easy_asc, n_easy);
    const int fg_idx = (n_bg > 0) ? vals[j] : rand_fg;
    const int bg_idx = ((j - fg_this) < hard_num) ? hp : ep;
    const int idx = (j < fg_this) ? fg_idx : bg_idx;

    const int gi = asgn_g[b * M_ + idx];
    const float iou_v = mo_s[idx];

    const float* rsrc = rois + ((size_t)b * M_ + idx) * 7;
    float* o_roi = out + OFF_ROIS + (size_t)(b * ROI_PER_IMAGE + j) * 7;
#pragma unroll
    for (int k = 0; k < 7; ++k) o_roi[k] = rsrc[k];

    const float* gsrc = gt + ((size_t)b * N_ + gi) * 8;
    float* o_gt = out + OFF_GT + (size_t)(b * ROI_PER_IMAGE + j) * 8;
#pragma unroll
    for (int k = 0; k < 8; ++k) o_gt[k] = gsrc[k];

    const int fo = b * ROI_PER_IMAGE + j;
    out[OFF_IOU + fo] = iou_v;
    out[OFF_SC + fo] = scores[b * M_ + idx];
    out[OFF_LB + fo] = (float)labels[b * M_ + idx];

    const bool fgc = iou_v > CLS_FG_THRESH;
    const bool bgc = iou_v < CLS_BG_THRESH;
    out[OFF_CLS + fo] = (!fgc && !bgc) ? (iou_v * 2.0f - 0.5f) : (fgc ? 1.0f : 0.0f);
    out[OFF_RV + fo] = (iou_v > REG_FG_THRESH) ? 1.0f : 0.0f;
  }
}

// ---------------- host entry ----------------
extern "C" void kernel_launch(void* const* d_in, const int* in_sizes, int n_in,
                              void* d_out, int out_size, void* d_ws, size_t ws_size,
                              hipStream_t stream) {
  (void)in_sizes; (void)n_in; (void)out_size; (void)ws_size;
  const float* rois   = (const float*)d_in[0];  // (B,M,7)
  const float* gt     = (const float*)d_in[1];  // (B,N,8)
  const float* scores = (const float*)d_in[2];  // (B,M)
  const int*   labels = (const int*)d_in[3];    // (B,M)
  float* out = (float*)d_out;

  float* mo = (float*)d_ws;                                   // B*M floats
  int* asgn = (int*)((char*)d_ws + (size_t)B_ * M_ * sizeof(float));  // B*M ints

  dim3 g1(B_, M_ / 128);
  rcnn_iou_kernel<<<g1, 128, 0, stream>>>(rois, gt, mo, asgn);
  rcnn_sample_kernel<<<B_, 256, 0, stream>>>(rois, gt, scores, labels, mo, asgn, out);
}